// CovPool_39805756899500
// MI455X (gfx1250) — compile-verified
//
#include <hip/hip_runtime.h>

typedef __attribute__((ext_vector_type(2))) float v2f;
typedef __attribute__((ext_vector_type(4))) float v4f;
typedef __attribute__((ext_vector_type(8))) float v8f;

#define BATCH   32
#define NROWS   16384
#define DDIM    64
#define SPLITS  16
#define ROWS_PER_BLOCK (NROWS / SPLITS)   // 1024
#define TILE_ROWS 64
#define LSTRIDE  66                        // 64 + 2: keeps b64 LDS reads 8B-aligned, spreads banks
#define DOUT     2080                      // 64*65/2
#define NTOT     (BATCH * DOUT)            // 66560
#define NPOW2    131072
#define CH       8192                      // LDS sort chunk (32 KB)
#define LAMBDA_  0.01f
#define FINF     3.402823466e38f

// wave -> (tile_i, tile_j, half) packed as 2-bit fields. 10 unique upper tiles:
// diagonals (0,0)(1,1)(2,2)(3,3) get one wave each (full k-range, half=2);
// the 6 off-diagonals get two waves each (half=0 -> rows 0..31, half=1 -> 32..63).
#define WI_PACK 0xA55000E4u
#define WJ_PACK 0xFFAFA5E4u
#define WH_PACK 0x444444AAu

// ---------------------------------------------------------------- workspace zero
__global__ void zero_ws_kernel(float* __restrict__ p, int n) {
  int i = blockIdx.x * blockDim.x + threadIdx.x;
  if (i < n) p[i] = 0.0f;
}

// ---------------------------------------------------------------- batched X^T X (upper tiles) + column sums
__global__ __launch_bounds__(512)
void cov_partial_kernel(const float* __restrict__ x,
                        float* __restrict__ S, float* __restrict__ sums) {
  __shared__ float lt[DDIM * LSTRIDE];   // transposed tile: lt[d][n]
  __shared__ float csum[DDIM];

  const int t     = threadIdx.x;
  const int b     = blockIdx.x >> 4;
  const int split = blockIdx.x & (SPLITS - 1);
  const float* xb = x + ((size_t)b * NROWS + (size_t)split * ROWS_PER_BLOCK) * DDIM;

  const int wave = t >> 5, lane = t & 31;
  const int di = (int)((WI_PACK >> (2 * wave)) & 3u) << 4;
  const int dj = (int)((WJ_PACK >> (2 * wave)) & 3u) << 4;
  const int wh = (int)((WH_PACK >> (2 * wave)) & 3u);   // 0,1 = half; 2 = full range
  const int m  = lane & 15;            // row within A tile / col within B tile
  const int k0 = (lane >> 4) << 1;     // lanes 0-15: K=0,1 ; lanes 16-31: K=2,3
  const int n0 = t >> 4;               // 0..31 (load row)
  const int d0 = (t << 2) & 63;        // 0..60 step 4 (load col group)

  v8f c = {};                          // 16x16 f32 accumulator (8 VGPRs)
  float a0 = 0.f, a1 = 0.f, a2 = 0.f, a3 = 0.f;
  const float* arow = &lt[(di + m) * LSTRIDE];
  const float* brow = &lt[(dj + m) * LSTRIDE];

  for (int tile = 0; tile < ROWS_PER_BLOCK / TILE_ROWS; ++tile) {
    const float* src = xb + tile * TILE_ROWS * DDIM;
    __syncthreads();                   // protect prior tile's LDS reads
#pragma unroll
    for (int rr = 0; rr < 2; ++rr) {   // each thread: two float4 = 8 elems, streamed NT
      const int n = n0 + rr * 32;
      const v4f v = __builtin_nontemporal_load((const v4f*)(src + n * DDIM + d0));
      lt[(d0 + 0) * LSTRIDE + n] = v[0];
      lt[(d0 + 1) * LSTRIDE + n] = v[1];
      lt[(d0 + 2) * LSTRIDE + n] = v[2];
      lt[(d0 + 3) * LSTRIDE + n] = v[3];
      a0 += v[0]; a1 += v[1]; a2 += v[2]; a3 += v[3];
    }
    __syncthreads();
#pragma unroll
    for (int h = 0; h < 2; ++h) {
      if (wh == 2 || wh == h) {        // wave-uniform branch; EXEC stays all-ones
#pragma unroll
        for (int n = h * 32; n < h * 32 + 32; n += 4) {
          // A[m][k0..k0+1] = X[n+k][di+m] ; B[k0..k0+1][m] = X[n+k][dj+m]
          v2f av = *(const v2f*)(arow + n + k0);
          v2f bv = *(const v2f*)(brow + n + k0);
          c = __builtin_amdgcn_wmma_f32_16x16x4_f32(
                  false, av, false, bv, (short)0, c, false, false);
        }
      }
    }
  }

  // -------- column-sum reduction (LDS then one global atomic per column)
  __syncthreads();
  if (t < DDIM) csum[t] = 0.0f;
  __syncthreads();
  atomicAdd(&csum[d0 + 0], a0);
  atomicAdd(&csum[d0 + 1], a1);
  atomicAdd(&csum[d0 + 2], a2);
  atomicAdd(&csum[d0 + 3], a3);
  __syncthreads();
  if (t < DDIM) atomicAdd(&sums[b * DDIM + t], csum[t]);

  // -------- spill S partials: C layout: lanes 0-15 -> M=r, lanes 16-31 -> M=8+r, N=lane&15
  const int hi = lane >> 4;
#pragma unroll
  for (int r = 0; r < 8; ++r) {
    const int row = di + hi * 8 + r;
    const int col = dj + m;
    atomicAdd(&S[((size_t)b * DDIM + row) * DDIM + col], c[r]);
  }
}

// ---------------------------------------------------------------- cov finalize + upper-tri extract
__global__ void finalize_kernel(const float* __restrict__ S,
                                const float* __restrict__ sums,
                                float* __restrict__ out) {
  const int b = blockIdx.x;
  const float invN   = 1.0f / (float)NROWS;
  const float invNm1 = 1.0f / (float)(NROWS - 1);
  const float* Sb = S + (size_t)b * DDIM * DDIM;
  const float* sb = sums + b * DDIM;
  for (int p = threadIdx.x; p < DOUT; p += blockDim.x) {
    int i = 0, rem = p;                 // invert triu flat index -> (i,j), j >= i
    while (rem >= DDIM - i) { rem -= DDIM - i; ++i; }
    const int j = i + rem;
    float cov = (Sb[i * DDIM + j] - sb[i] * sb[j] * invN) * invNm1;
    if (i == j) cov += LAMBDA_;
    out[b * DOUT + p] = cov;
  }
}

// ---------------------------------------------------------------- ascending-only bitonic, global strides
// j==0 => mirror phase of merge size k; else strided phase j. Comparators with the
// high index >= NTOT are skipped (virtual +INF padding never moves down).
__global__ void bitonic_global_kernel(float* __restrict__ a, int k, int j) {
  const int t = blockIdx.x * blockDim.x + threadIdx.x;   // 0 .. NPOW2/2-1
  int i, p;
  if (j == 0) {
    const int half = k >> 1;
    const int blk  = (t / half) * k;
    const int off  = t & (half - 1);
    i = blk + off;
    p = blk + (k - 1) - off;
  } else {
    i = ((t & ~(j - 1)) << 1) | (t & (j - 1));
    p = i + j;
  }
  if (p < NTOT) {
    const float vi = a[i], vp = a[p];
    if (vp < vi) { a[i] = vp; a[p] = vi; }
  }
}

// ---------------------------------------------------------------- full sort of one 8192 chunk in LDS
__global__ __launch_bounds__(1024)
void bitonic_local_sort_kernel(float* __restrict__ a) {
  __shared__ float s[CH];
  const int base = blockIdx.x * CH;
  if (base >= NTOT) return;
  const int t = threadIdx.x;
  for (int e = t; e < CH; e += 1024) {
    const int gi = base + e;
    s[e] = (gi < NTOT) ? a[gi] : FINF;
  }
  __syncthreads();
  for (int k = 2; k <= CH; k <<= 1) {
    const int half = k >> 1;
    for (int cmp = t; cmp < CH / 2; cmp += 1024) {      // mirror phase
      const int blk = (cmp / half) * k;
      const int off = cmp & (half - 1);
      const int i = blk + off, p = blk + (k - 1) - off;
      const float vi = s[i], vp = s[p];
      if (vp < vi) { s[i] = vp; s[p] = vi; }
    }
    __syncthreads();
    for (int j = k >> 2; j >= 1; j >>= 1) {             // strided phases
      for (int cmp = t; cmp < CH / 2; cmp += 1024) {
        const int i = ((cmp & ~(j - 1)) << 1) | (cmp & (j - 1));
        const int p = i + j;
        const float vi = s[i], vp = s[p];
        if (vp < vi) { s[i] = vp; s[p] = vi; }
      }
      __syncthreads();
    }
  }
  for (int e = t; e < CH; e += 1024) {
    const int gi = base + e;
    if (gi < NTOT) a[gi] = s[e];
  }
}

// ---------------------------------------------------------------- in-chunk merge tail: strides CH/2..1
__global__ __launch_bounds__(1024)
void bitonic_local_merge_kernel(float* __restrict__ a) {
  __shared__ float s[CH];
  const int base = blockIdx.x * CH;
  if (base >= NTOT) return;
  const int t = threadIdx.x;
  for (int e = t; e < CH; e += 1024) {
    const int gi = base + e;
    s[e] = (gi < NTOT) ? a[gi] : FINF;
  }
  __syncthreads();
  for (int j = CH >> 1; j >= 1; j >>= 1) {
    for (int cmp = t; cmp < CH / 2; cmp += 1024) {
      const int i = ((cmp & ~(j - 1)) << 1) | (cmp & (j - 1));
      const int p = i + j;
      const float vi = s[i], vp = s[p];
      if (vp < vi) { s[i] = vp; s[p] = vi; }
    }
    __syncthreads();
  }
  for (int e = t; e < CH; e += 1024) {
    const int gi = base + e;
    if (gi < NTOT) a[gi] = s[e];
  }
}

// ---------------------------------------------------------------- launcher
extern "C" void kernel_launch(void* const* d_in, const int* in_sizes, int n_in,
                              void* d_out, int out_size, void* d_ws, size_t ws_size,
                              hipStream_t stream) {
  (void)in_sizes; (void)n_in; (void)out_size; (void)ws_size;
  const float* x = (const float*)d_in[0];
  float* out  = (float*)d_out;                    // 66560 f32, sorted in place
  float* S    = (float*)d_ws;                     // 32*64*64 partial Gram (upper tiles)
  float* sums = S + BATCH * DDIM * DDIM;          // 32*64 column sums

  const int nzero = BATCH * DDIM * DDIM + BATCH * DDIM;
  zero_ws_kernel<<<(nzero + 255) / 256, 256, 0, stream>>>(S, nzero);

  cov_partial_kernel<<<BATCH * SPLITS, 512, 0, stream>>>(x, S, sums);
  finalize_kernel<<<BATCH, 256, 0, stream>>>(S, sums, out);

  // hybrid ascending-only bitonic sort of 66560 values (virtual pad to 131072)
  bitonic_local_sort_kernel<<<NPOW2 / CH, 1024, 0, stream>>>(out);
  for (int k = CH * 2; k <= NPOW2; k <<= 1) {
    bitonic_global_kernel<<<NPOW2 / 2 / 256, 256, 0, stream>>>(out, k, 0);   // mirror
    for (int j = k >> 2; j >= CH; j >>= 1)
      bitonic_global_kernel<<<NPOW2 / 2 / 256, 256, 0, stream>>>(out, k, j); // cross-chunk strides
    bitonic_local_merge_kernel<<<NPOW2 / CH, 1024, 0, stream>>>(out);        // strides 4096..1 in LDS
  }
}